// UpSampleWave_28037546508457
// MI455X (gfx1250) — compile-verified
//
#include <hip/hip_runtime.h>

// UpSampleWave: zero-insert upsample x2 along W + depthwise 1x4 blur [1,3,3,1]/8.
// Pure bandwidth kernel (0.33 FLOP/byte): 50MB in, 100MB out -> ~6.5us at 23.3TB/s.
// CDNA5 path used: global_load_async_to_lds_b128 (ASYNCcnt) staging + LDS halo
// resolution + non-temporal b128 stores.

typedef float v4f __attribute__((ext_vector_type(4)));

#define BLOCK   256
#define TILE_IN 1024   // input floats per block (BLOCK lanes x 16B)

__global__ __launch_bounds__(BLOCK)
void upsample_wave_kernel(const float* __restrict__ x,
                          const float* __restrict__ blur,   // (3,1,1,4) taps
                          float* __restrict__ out,
                          int W)                            // input row width
{
    __shared__ __align__(16) float smem[TILE_IN];

    const int tid       = threadIdx.x;
    const int tile      = blockIdx.x;          // tile index within row
    const int row       = blockIdx.y;          // 0 .. B*C-1
    const int tileStart = tile * TILE_IN;

    const long long rowInBase  = (long long)row * (long long)W;
    const long long rowOutBase = 2ll * rowInBase;

    // ---- Async global -> LDS stage: each lane moves 16 bytes -----------------
    {
        const float* gsrc = x + rowInBase + tileStart + (tid << 2);
        unsigned long long gaddr = (unsigned long long)gsrc;
        // low 32 bits of a flat shared pointer == LDS byte address
        unsigned ldsAddr = (unsigned)(unsigned long long)(const void*)(&smem[tid << 2]);
        asm volatile(
            "global_load_async_to_lds_b128 %0, %1, off\n"
            :
            : "v"(ldsAddr), "v"(gaddr)
            : "memory");
    }
    asm volatile("s_wait_asynccnt 0" ::: "memory");
    __syncthreads();

    // ---- Gather 4 samples + halo from LDS ------------------------------------
    const int j0 = tid << 2;
    const float a0 = smem[j0 + 0];
    const float a1 = smem[j0 + 1];
    const float a2 = smem[j0 + 2];
    const float a3 = smem[j0 + 3];

    float xl, xr;
    if (tid > 0) {
        xl = smem[j0 - 1];
    } else {
        xl = (tileStart > 0) ? x[rowInBase + tileStart - 1] : 0.0f;
    }
    if (tid < BLOCK - 1) {
        xr = smem[j0 + 4];
    } else {
        xr = (tileStart + TILE_IN < W) ? x[rowInBase + tileStart + TILE_IN] : 0.0f;
    }

    // ---- Filter taps (per channel; uniform across block) ---------------------
    const int  c  = row % 3;                 // row = b*C + c
    const float* k = blur + c * 4;
    const float k0 = k[0], k1 = k[1], k2 = k[2], k3 = k[3];

    // out[2j]   = k0*x[j-1] + k2*x[j]
    // out[2j+1] = k1*x[j]   + k3*x[j+1]
    v4f o0, o1;
    o0.x = k0 * xl + k2 * a0;
    o0.y = k1 * a0 + k3 * a1;
    o0.z = k0 * a0 + k2 * a1;
    o0.w = k1 * a1 + k3 * a2;
    o1.x = k0 * a1 + k2 * a2;
    o1.y = k1 * a2 + k3 * a3;
    o1.z = k0 * a2 + k2 * a3;
    o1.w = k1 * a3 + k3 * xr;

    // ---- Stream out: 8 contiguous floats per thread, non-temporal b128 -------
    v4f* dst = (v4f*)(out + rowOutBase + 2ll * (long long)tileStart + (tid << 3));
    __builtin_nontemporal_store(o0, dst + 0);
    __builtin_nontemporal_store(o1, dst + 1);
}

extern "C" void kernel_launch(void* const* d_in, const int* in_sizes, int n_in,
                              void* d_out, int out_size, void* d_ws, size_t ws_size,
                              hipStream_t stream) {
    const float* x    = (const float*)d_in[0];   // (16,3,1,262144) f32
    const float* blur = (const float*)d_in[1];   // (3,1,1,4) f32
    float*       out  = (float*)d_out;           // (16,3,1,524288) f32

    const int W    = 262144;
    const int rows = in_sizes[0] / W;            // B*C = 48

    dim3 grid(W / TILE_IN, rows);                // (256, 48)
    upsample_wave_kernel<<<grid, BLOCK, 0, stream>>>(x, blur, out, W);
}